// VGNeRF_2396591751241
// MI455X (gfx1250) — compile-verified
//
#include <hip/hip_runtime.h>
#include <hip/hip_bf16.h>
#include <math.h>

typedef float v2f __attribute__((ext_vector_type(2)));
typedef float v8f __attribute__((ext_vector_type(8)));

#define N_PTS 1024
#define IMG_W 128
#define IMG_H 128
#define NEARP 0.01f
#define NSIG2 9.0f      // NSIGMA^2
#define COVBLUR 0.3f

// ---------------------------------------------------------------------------
// Kernel 1: per-point camera transform + 2D conic -> quadratic-form
// coefficients, bitonic depth sort (LDS), scatter to sorted workspace.
//   coefS[rank][8] = {ca, cc, 2cb, -2(ca*u+cb*v), -2(cc*v+cb*u), const, 0, 0}
//   pcolS[rank]    = float4{r, g, b, alpha (0 if invalid)}
// ---------------------------------------------------------------------------
__global__ __launch_bounds__(1024) void gs_preprocess(
    const float* __restrict__ coords, const float* __restrict__ cov,
    const float* __restrict__ colors, const float* __restrict__ alphas,
    const float* __restrict__ Wm, const float* __restrict__ Km,
    float* __restrict__ coefS, float4* __restrict__ pcolS)
{
  __shared__ float skey[N_PTS];
  __shared__ int   sidx[N_PTS];
  __shared__ int   sinv[N_PTS];
  const int t = threadIdx.x;

  float R[3][3], tv[3];
#pragma unroll
  for (int i = 0; i < 3; ++i) {
#pragma unroll
    for (int j = 0; j < 3; ++j) R[i][j] = Wm[4 * i + j];
    tv[i] = Wm[4 * i + 3];
  }
  const float fx = Km[0], fy = Km[4], cx = Km[2], cy = Km[5];

  const float px = coords[3 * t + 0];
  const float py = coords[3 * t + 1];
  const float pz = coords[3 * t + 2];
  const float tx = R[0][0] * px + R[0][1] * py + R[0][2] * pz + tv[0];
  const float ty = R[1][0] * px + R[1][1] * py + R[1][2] * pz + tv[1];
  const float tz = R[2][0] * px + R[2][1] * py + R[2][2] * pz + tv[2];

  const float iz = 1.0f / fmaxf(tz, NEARP);
  const float u = fx * tx * iz + cx;
  const float v = fy * ty * iz + cy;

  // sig_cam = R * Sigma * R^T
  const float* S = cov + 9 * t;
  float Mt[3][3], SC[3][3];
#pragma unroll
  for (int i = 0; i < 3; ++i)
#pragma unroll
    for (int k = 0; k < 3; ++k)
      Mt[i][k] = R[i][0] * S[0 * 3 + k] + R[i][1] * S[1 * 3 + k] + R[i][2] * S[2 * 3 + k];
#pragma unroll
  for (int i = 0; i < 3; ++i)
#pragma unroll
    for (int l = 0; l < 3; ++l)
      SC[i][l] = Mt[i][0] * R[l][0] + Mt[i][1] * R[l][1] + Mt[i][2] * R[l][2];

  // sig2d = J * sig_cam * J^T + blur*I   (J is 2x3 with zeros in [0][1],[1][0])
  const float iz2 = iz * iz;
  const float J00 = fx * iz, J02 = -fx * tx * iz2;
  const float J11 = fy * iz, J12 = -fy * ty * iz2;
  const float JS00 = J00 * SC[0][0] + J02 * SC[2][0];
  const float JS01 = J00 * SC[0][1] + J02 * SC[2][1];
  const float JS02 = J00 * SC[0][2] + J02 * SC[2][2];
  const float JS11 = J11 * SC[1][1] + J12 * SC[2][1];
  const float JS12 = J11 * SC[1][2] + J12 * SC[2][2];
  const float s00 = JS00 * J00 + JS02 * J02 + COVBLUR;
  const float s01 = JS01 * J11 + JS02 * J12;
  const float s11 = JS11 * J11 + JS12 * J12 + COVBLUR;

  const float det = s00 * s11 - s01 * s01;
  const float idet = 1.0f / det;
  const float ca = s11 * idet;
  const float cc = s00 * idet;
  const float cb = -s01 * idet;

  float a = fminf(fmaxf(alphas[t], 0.0f), 1.0f);
  if (!(tz > NEARP)) a = 0.0f;  // fold validity into alpha

  // q(x,y) = c0*x^2 + c1*y^2 + c2*x*y + c3*x + c4*y + c5
  const float c0 = ca, c1 = cc, c2 = 2.0f * cb;
  const float c3 = -2.0f * (ca * u + cb * v);
  const float c4 = -2.0f * (cc * v + cb * u);
  const float c5 = ca * u * u + cc * v * v + 2.0f * cb * u * v;

  // bitonic sort (key=tz ascending, val=point index)
  skey[t] = tz;
  sidx[t] = t;
  __syncthreads();
  for (int k = 2; k <= N_PTS; k <<= 1) {
    for (int j = k >> 1; j > 0; j >>= 1) {
      const int ixj = t ^ j;
      if (ixj > t) {
        const float a1 = skey[t], a2 = skey[ixj];
        const bool up = ((t & k) == 0);
        if ((a1 > a2) == up) {
          skey[t] = a2; skey[ixj] = a1;
          const int i1 = sidx[t]; sidx[t] = sidx[ixj]; sidx[ixj] = i1;
        }
      }
      __syncthreads();
    }
  }
  sinv[sidx[t]] = t;   // inverse permutation: original -> rank
  __syncthreads();

  const int rank = sinv[t];
  float* cw = coefS + (size_t)rank * 8;
  cw[0] = c0; cw[1] = c1; cw[2] = c2; cw[3] = c3;
  cw[4] = c4; cw[5] = c5; cw[6] = 0.0f; cw[7] = 0.0f;
  pcolS[rank] = make_float4(colors[3 * t], colors[3 * t + 1], colors[3 * t + 2], a);
}

// ---------------------------------------------------------------------------
// Kernel 2: rasterize. Each wave owns 32 pixels of one row. Per 16-point
// chunk: q-tiles via chained V_WMMA_F32_16X16X4_F32 (K=6 padded to 8), then
// sequential front-to-back alpha blend. Point color/alpha is read with a
// wave-uniform index -> scalar s_load path (constant cache broadcast), no LDS
// staging needed. A-matrix loads are double-buffered one chunk ahead and the
// coefficient stream is prefetched two chunks ahead (global_prefetch_b8).
// ---------------------------------------------------------------------------
__global__ __launch_bounds__(256) void gs_raster(
    const float* __restrict__ coefS, const float4* __restrict__ pcolS,
    float* __restrict__ out)
{
  const int lane = threadIdx.x & 31;
  const int wave = threadIdx.x >> 5;
  const int pixBase = blockIdx.x * 256 + wave * 32;   // 32 pixels per wave
  const int y  = pixBase >> 7;       // IMG_W == 128
  const int x0 = pixBase & 127;
  const int pl = lane & 15;
  const int kh = lane >> 4;          // which K-half this lane carries
  const bool hi = (lane >= 16);

  // Loop-invariant B operands: per-pixel monomials [x^2, y^2, xy, x, y, 1, 0, 0]
  const float yb  = (float)y + 0.5f;
  const float xb0 = (float)(x0 + pl) + 0.5f;   // tile0 pixels x0..x0+15
  const float xb1 = xb0 + 16.0f;               // tile1 pixels x0+16..x0+31
  const float m0[8] = {xb0 * xb0, yb * yb, xb0 * yb, xb0, yb, 1.0f, 0.0f, 0.0f};
  const float m1[8] = {xb1 * xb1, yb * yb, xb1 * yb, xb1, yb, 1.0f, 0.0f, 0.0f};
  const v2f B0lo = {m0[2 * kh], m0[2 * kh + 1]};
  const v2f B0hi = {m0[4 + 2 * kh], m0[5 + 2 * kh]};
  const v2f B1lo = {m1[2 * kh], m1[2 * kh + 1]};
  const v2f B1hi = {m1[4 + 2 * kh], m1[5 + 2 * kh]};

  float T = 1.0f, accr = 0.0f, accg = 0.0f, accb = 0.0f;

  // A operand: 16x4 f32 layout — lane m (0..15) carries row M=m, K={0,1};
  // lane m+16 carries row M=m, K={2,3}. Second WMMA covers K={4..7}.
  const v2f* ap0 = (const v2f*)(coefS + (size_t)pl * 8);
  v2f Alo = ap0[kh];
  v2f Ahi = ap0[2 + kh];

  for (int n = 0; n < N_PTS / 16; ++n) {
    const int base = n * 16;

    // Prefetch coefficient stream two chunks ahead (gfx1250 global_prefetch_b8).
    __builtin_prefetch(coefS + (size_t)(((n + 2) & 63) * 16 + pl) * 8, 0, 1);

    v8f D0 = {}; v8f D1 = {};
    D0 = __builtin_amdgcn_wmma_f32_16x16x4_f32(false, Alo, false, B0lo, (short)0, D0, false, false);
    D0 = __builtin_amdgcn_wmma_f32_16x16x4_f32(false, Ahi, false, B0hi, (short)0, D0, false, false);
    D1 = __builtin_amdgcn_wmma_f32_16x16x4_f32(false, Alo, false, B1lo, (short)0, D1, false, false);
    D1 = __builtin_amdgcn_wmma_f32_16x16x4_f32(false, Ahi, false, B1hi, (short)0, D1, false, false);

    // Double-buffer: issue next chunk's A loads before the sequential blend.
    {
      const v2f* apn = (const v2f*)(coefS + (size_t)(((n + 1) & 63) * 16 + pl) * 8);
      Alo = apn[kh];
      Ahi = apn[2 + kh];
    }

    // C/D layout: VGPR r = {M=r for lanes 0..15, M=r+8 for lanes 16..31}.
    // Exchange halves so each lane holds all 16 q's of its own pixel in order.
    float myq[16];
#pragma unroll
    for (int r = 0; r < 8; ++r) {
      const float d0 = D0[r], d1 = D1[r];
      const float o0 = __shfl_xor(d0, 16, 32);
      const float o1 = __shfl_xor(d1, 16, 32);
      myq[r]     = hi ? o1 : d0;
      myq[r + 8] = hi ? d1 : o0;
    }

    // Sequential front-to-back blend over this chunk's 16 points.
    // base+j is wave-uniform -> backend scalarizes to s_load_b128 broadcast.
#pragma unroll
    for (int j = 0; j < 16; ++j) {
      const float4 pc = pcolS[base + j];
      const float q = myq[j];
      const float g = __expf(-0.5f * q);
      float apx = (q <= NSIG2) ? pc.w * g : 0.0f;
      apx = fminf(apx, 0.999f);
      const float w = apx * T;
      accr += w * pc.x;
      accg += w * pc.y;
      accb += w * pc.z;
      T *= (1.0f - apx);
    }
  }

  const int pix = y * IMG_W + x0 + lane;
  out[pix] = accr;
  out[IMG_H * IMG_W + pix] = accg;
  out[2 * IMG_H * IMG_W + pix] = accb;
}

extern "C" void kernel_launch(void* const* d_in, const int* in_sizes, int n_in,
                              void* d_out, int out_size, void* d_ws, size_t ws_size,
                              hipStream_t stream) {
  const float* coords = (const float*)d_in[0];
  const float* covs   = (const float*)d_in[1];
  const float* colors = (const float*)d_in[2];
  const float* alphas = (const float*)d_in[3];
  const float* Wm     = (const float*)d_in[4];
  const float* Km     = (const float*)d_in[5];

  float*  coefS = (float*)d_ws;                       // 1024*8 f32 = 32 KB
  float4* pcolS = (float4*)((char*)d_ws + N_PTS * 8 * sizeof(float)); // 16 KB

  gs_preprocess<<<1, 1024, 0, stream>>>(coords, covs, colors, alphas, Wm, Km,
                                        coefS, pcolS);
  gs_raster<<<(IMG_H * IMG_W) / 256, 256, 0, stream>>>(coefS, pcolS,
                                                       (float*)d_out);
}